// SPDTangentSpace1_40338332844038
// MI455X (gfx1250) — compile-verified
//
#include <hip/hip_runtime.h>

// ---------------------------------------------------------------------------
// Batched SPD matrix-log for 8192 x (64x64) via inverse scaling-and-squaring:
//   B = A/s  (s = ||A||_F)                     -> eigenvalues in (0,1]
//   repeat NSQRT times: B <- sqrt(B)           (Newton-Schulz, GEMM-only)
//   log(A) = 2^NSQRT * log(I + (B - I)) + log(s)*I   (Taylor/Horner, GEMMs)
// All GEMMs are 64x64x64 fp32 on V_WMMA_F32_16X16X4_F32 (CDNA5 wave32 WMMA).
// ---------------------------------------------------------------------------

typedef __attribute__((ext_vector_type(2))) float v2f;
typedef __attribute__((ext_vector_type(8))) float v8f;

#if defined(__HIP_DEVICE_COMPILE__)
#if !__has_builtin(__builtin_amdgcn_wmma_f32_16x16x4_f32)
#error "device pass: missing __builtin_amdgcn_wmma_f32_16x16x4_f32"
#endif
#endif

#define NSQRT  6    // number of matrix square roots
#define NITER  14   // Newton-Schulz iterations per sqrt
#define NTERMS 8    // Taylor terms for log(I+E)
#define TRI    2080 // 64*65/2 upper-tri entries

// Rotate-swizzled LDS layout: element (r,c) at r*64 + ((c+r)&63).
// Row access (fixed r, varying c) and column access (fixed c, varying r)
// are both bank-conflict-free across the 64 LDS banks -> no padding needed,
// 4 buffers fit in exactly 64 KB.
__device__ __forceinline__ int sidx(int r, int c) {
  return (r << 6) + ((c + r) & 63);
}

// One 64x64x64 fp32 GEMM: C = A * B   (WMODE=false)
//                         C = A * (1.5*I - 0.5*B)   (WMODE=true, NS W fused)
// 8 waves; wave w computes output tiles (w>>1, (w&1)*2) and (w>>1, (w&1)*2+1),
// sharing the A operand between the two tiles.
template <bool WMODE>
__device__ __forceinline__ void gemm64(const float* __restrict__ A,
                                       const float* __restrict__ B,
                                       float* __restrict__ C,
                                       int lane, int wave) {
  const int mt   = wave >> 1;         // 0..3   output row tile
  const int nt0  = (wave & 1) * 2;    // 0 or 2 first output col tile
  const int arow = mt * 16 + (lane & 15);
  const int koff = (lane >> 4) * 2;   // ISA: lanes 0-15 hold K,K+1; 16-31 hold K+2,K+3
  const int col0 = nt0 * 16 + (lane & 15);
  const int col1 = col0 + 16;

  v8f acc0 = {};
  v8f acc1 = {};
#pragma unroll
  for (int k0 = 0; k0 < 64; k0 += 4) {
    const int ka = k0 + koff;
    v2f a;
    a.x = A[sidx(arow, ka)];
    a.y = A[sidx(arow, ka + 1)];
    v2f b0, b1;
    if (WMODE) {
      b0.x = (ka     == col0 ? 1.5f : 0.0f) - 0.5f * B[sidx(ka,     col0)];
      b0.y = (ka + 1 == col0 ? 1.5f : 0.0f) - 0.5f * B[sidx(ka + 1, col0)];
      b1.x = (ka     == col1 ? 1.5f : 0.0f) - 0.5f * B[sidx(ka,     col1)];
      b1.y = (ka + 1 == col1 ? 1.5f : 0.0f) - 0.5f * B[sidx(ka + 1, col1)];
    } else {
      b0.x = B[sidx(ka,     col0)];
      b0.y = B[sidx(ka + 1, col0)];
      b1.x = B[sidx(ka,     col1)];
      b1.y = B[sidx(ka + 1, col1)];
    }
#if defined(__HIP_DEVICE_COMPILE__)
    // (neg_a, A, neg_b, B, c_mod, C, reuse_a, reuse_b)
    acc0 = __builtin_amdgcn_wmma_f32_16x16x4_f32(false, a, false, b0,
                                                 (short)0, acc0, false, false);
    acc1 = __builtin_amdgcn_wmma_f32_16x16x4_f32(false, a, false, b1,
                                                 (short)0, acc1, false, false);
#else
    (void)a; (void)b0; (void)b1;
#endif
  }
  // C layout: VGPR r -> row (mt*16 + r) for lanes 0-15, (mt*16 + 8 + r) for 16-31
  const int crow  = mt * 16 + (lane >> 4) * 8;
  const int ccol0 = nt0 * 16 + (lane & 15);
#pragma unroll
  for (int r = 0; r < 8; ++r) {
    C[sidx(crow + r, ccol0)]      = acc0[r];
    C[sidx(crow + r, ccol0 + 16)] = acc1[r];
  }
}

extern "C" __global__ void __launch_bounds__(256)
spd_logm_kernel(const float* __restrict__ x, float* __restrict__ out) {
  __shared__ float smem[4 * 64 * 64];  // exactly 64 KB
  float* buf0 = smem;
  float* buf1 = smem + 4096;
  float* buf2 = smem + 8192;
  float* buf3 = smem + 12288;

  const int tid  = threadIdx.x;
  const int lane = tid & 31;
  const int wave = tid >> 5;
  const long long b = blockIdx.x;
  const float* __restrict__ xb = x + b * 4096ll;

  // ---- load A (swizzled) + Frobenius-norm partial sums ----
  float ps = 0.0f;
#pragma unroll
  for (int i = tid; i < 4096; i += 256) {
    const float v = xb[i];
    buf0[sidx(i >> 6, i & 63)] = v;
    ps += v * v;
  }
  buf2[tid] = ps;
  __syncthreads();
  for (int off = 128; off > 0; off >>= 1) {
    if (tid < off) buf2[tid] += buf2[tid + off];
    __syncthreads();
  }
  const float fro  = sqrtf(buf2[0]);
  const float invs = 1.0f / fro;
  __syncthreads();

  // ---- Y = A / s ----
  for (int i = tid; i < 4096; i += 256) buf0[i] *= invs;

  // ---- NSQRT repeated square roots via Newton-Schulz product form ----
  // Y' = Y*W,  M' = (M*W)*W,  W = 1.5I - 0.5M  (fused into B-operand load).
  float* pY = buf0;
  float* pM = buf1;
  float* pA = buf2;  // scratch (gets Y')
  float* pB = buf3;  // scratch (gets M*W)
  for (int sq = 0; sq < NSQRT; ++sq) {
    __syncthreads();
    for (int i = tid; i < 4096; i += 256) pM[i] = pY[i];  // M0 = Y
    for (int it = 0; it < NITER; ++it) {
      __syncthreads();
      gemm64<true>(pY, pM, pA, lane, wave);   // Y'  = Y * W(M)
      gemm64<true>(pM, pM, pB, lane, wave);   // MW  = M * W(M)
      __syncthreads();
      gemm64<true>(pB, pM, pY, lane, wave);   // M'  = MW * W(M) -> old Y buf
      float* nY = pA; float* nM = pY; float* nA = pM; float* nB = pB;
      pY = nY; pM = nM; pA = nA; pB = nB;
    }
  }

  // ---- E = Y - I ----
  __syncthreads();
  float* pE = pA;
  for (int i = tid; i < 4096; i += 256) {
    const int r = i >> 6;
    const float d = ((i & 63) == ((r + r) & 63)) ? 1.0f : 0.0f;
    pE[i] = pY[i] - d;
  }

  // ---- Horner for log(I+E) = E*(I - E*(I/2 - E*(I/3 - ...))) ----
  float* pS = pB;
  float* pT = pY;  // Y no longer needed
  __syncthreads();
  for (int i = tid; i < 4096; i += 256) {
    const int r = i >> 6;
    pS[i] = ((i & 63) == ((r + r) & 63)) ? (1.0f / (float)NTERMS) : 0.0f;
  }
  for (int t = NTERMS - 1; t >= 1; --t) {
    __syncthreads();
    gemm64<false>(pE, pS, pT, lane, wave);    // T = E * S
    __syncthreads();
    const float ct = 1.0f / (float)t;
    for (int i = tid; i < 4096; i += 256) {
      const int r = i >> 6;
      const float d = ((i & 63) == ((r + r) & 63)) ? ct : 0.0f;
      pT[i] = d - pT[i];                      // S_t = I/t - E*S_{t+1}
    }
    float* tmp = pS; pS = pT; pT = tmp;
  }
  __syncthreads();
  gemm64<false>(pE, pS, pT, lane, wave);      // G = E * S_1
  __syncthreads();

  // ---- out = 2^NSQRT * G + log(s)*I, upper triangle row-major ----
  const float scale2 = (float)(1 << NSQRT);
  const float logs   = logf(fro);
  for (int i = tid; i < TRI; i += 256) {
    int r = 0, rem = i;
    while (rem >= (64 - r)) { rem -= (64 - r); ++r; }
    const int c = r + rem;
    const float v = scale2 * pT[sidx(r, c)] + (r == c ? logs : 0.0f);
    out[b * (long long)TRI + i] = v;
  }
}

extern "C" void kernel_launch(void* const* d_in, const int* in_sizes, int n_in,
                              void* d_out, int out_size, void* d_ws, size_t ws_size,
                              hipStream_t stream) {
  const float* x = (const float*)d_in[0];
  float* out = (float*)d_out;
  const int B = in_sizes[0] / 4096;  // 8192 matrices of 64x64
  spd_logm_kernel<<<dim3(B), dim3(256), 0, stream>>>(x, out);
}